// MaxSupPixPool_17179869891
// MI455X (gfx1250) — compile-verified
//
#include <hip/hip_runtime.h>
#include <hip/hip_bf16.h>
#include <stdint.h>

#define KSEG   1024
#define CCH    128
#define HWPIX  (512 * 512)
#define NBATCH 4
#define CBLK   8        // channels per workgroup
#define PIXWG  8192     // pixels per workgroup
#define PTILES (HWPIX / PIXWG)   // 32
#define ENC_NEG_INF 0x007FFFFFu  // encode(-inf)

typedef float __attribute__((ext_vector_type(4))) nt_float4;  // native clang vector
typedef int   __attribute__((ext_vector_type(4))) nt_int4;

// Order-preserving float->uint map: f >= g  <=>  enc(f) >= enc(g) (unsigned)
__device__ __forceinline__ unsigned enc_f32(float f) {
    unsigned u = __float_as_uint(f);
    return (u & 0x80000000u) ? ~u : (u | 0x80000000u);
}
__device__ __forceinline__ float dec_f32(unsigned u) {
    unsigned v = (u & 0x80000000u) ? (u & 0x7FFFFFFFu) : ~u;
    return __uint_as_float(v);
}

__global__ void spx_init_out(unsigned* __restrict__ out, int n) {
    int i = blockIdx.x * blockDim.x + threadIdx.x;
    if (i < n) out[i] = ENC_NEG_INF;
}

__global__ void spx_decode_out(float* __restrict__ out, int n) {
    int i = blockIdx.x * blockDim.x + threadIdx.x;
    if (i < n) {
        unsigned u = ((const unsigned*)out)[i];
        out[i] = dec_f32(u);
    }
}

// grid = (PTILES, CCH/CBLK, NBATCH), block = 256
__global__ void __launch_bounds__(256)
spx_segmax_main(const float* __restrict__ img,
                const int*   __restrict__ spx,
                unsigned*    __restrict__ out) {
    __shared__ __align__(16) float segf[CBLK * KSEG];     // 32 KB float maxima
    __shared__ __align__(16) int   labels[PIXWG];         // 32 KB label tile

    const int tid  = threadIdx.x;
    const int pix0 = blockIdx.x * PIXWG;
    const int cb   = blockIdx.y * CBLK;
    const int b    = blockIdx.z;

    // ---- Stage the label tile into LDS with the CDNA5 async data mover ----
    {
        const int* gsrc = spx + (size_t)b * HWPIX + pix0;
        #pragma unroll
        for (int j = 0; j < PIXWG / (256 * 4); ++j) {
            int e = (j * 256 + tid) * 4;                   // int index, 16B aligned
            unsigned lds_off = (unsigned)(uintptr_t)(&labels[e]);
            uint64_t gaddr   = (uint64_t)(uintptr_t)(gsrc + e);
            asm volatile("global_load_async_to_lds_b128 %0, %1, off"
                         :: "v"(lds_off), "v"(gaddr) : "memory");
        }
    }

    // ---- Overlap: init the segment-max table while the async copy flies ----
    #pragma unroll
    for (int e = tid; e < CBLK * KSEG; e += 256) segf[e] = -INFINITY;

    __builtin_amdgcn_s_wait_asynccnt(0);
    __syncthreads();

    // ---- Main reduction: NT float4 img reads, ds_max_num_f32 scatter ----
    const nt_int4* lab4 = (const nt_int4*)labels;
    #pragma unroll
    for (int i = 0; i < PIXWG / (256 * 4); ++i) {          // 8 iterations
        const int      idx4 = i * 256 + tid;
        const nt_int4  lab  = lab4[idx4];
        const int      p    = pix0 + idx4 * 4;

        // Per-pixel LDS addresses hoisted; channel row folds into DS imm offset.
        float* a0 = &segf[lab.x];
        float* a1 = &segf[lab.y];
        float* a2 = &segf[lab.z];
        float* a3 = &segf[lab.w];
        const float* ip = img + ((size_t)(b * CCH + cb)) * HWPIX + p;

        #pragma unroll
        for (int c = 0; c < CBLK; ++c) {
            const nt_float4 v = __builtin_nontemporal_load(
                (const nt_float4*)(ip + (size_t)c * HWPIX));
            __hip_atomic_fetch_max(a0 + c * KSEG, v.x,
                __ATOMIC_RELAXED, __HIP_MEMORY_SCOPE_WORKGROUP);
            __hip_atomic_fetch_max(a1 + c * KSEG, v.y,
                __ATOMIC_RELAXED, __HIP_MEMORY_SCOPE_WORKGROUP);
            __hip_atomic_fetch_max(a2 + c * KSEG, v.z,
                __ATOMIC_RELAXED, __HIP_MEMORY_SCOPE_WORKGROUP);
            __hip_atomic_fetch_max(a3 + c * KSEG, v.w,
                __ATOMIC_RELAXED, __HIP_MEMORY_SCOPE_WORKGROUP);
        }
    }

    __syncthreads();

    // ---- Merge: encode once per table entry, u32 atomic max to global ----
    #pragma unroll
    for (int e = tid; e < CBLK * KSEG; e += 256) {
        const int c = e >> 10;
        const int k = e & (KSEG - 1);
        atomicMax(out + ((size_t)(b * CCH + cb + c)) * KSEG + k, enc_f32(segf[e]));
    }
}

extern "C" void kernel_launch(void* const* d_in, const int* in_sizes, int n_in,
                              void* d_out, int out_size, void* d_ws, size_t ws_size,
                              hipStream_t stream) {
    const float* img = (const float*)d_in[0];
    const int*   spx = (const int*)d_in[1];
    unsigned*    outu = (unsigned*)d_out;

    const int n = NBATCH * CCH * KSEG;   // 524288 == out_size

    spx_init_out<<<n / 256, 256, 0, stream>>>(outu, n);

    dim3 grid(PTILES, CCH / CBLK, NBATCH);
    spx_segmax_main<<<grid, 256, 0, stream>>>(img, spx, outu);

    spx_decode_out<<<n / 256, 256, 0, stream>>>((float*)d_out, n);
}